// DistanceWeightedLoss_78597901517283
// MI455X (gfx1250) — compile-verified
//
#include <hip/hip_runtime.h>

// Problem constants (fixed by the reference setup)
#define B_   16
#define H_   256
#define W_   256
#define INF_ 1.0e6f
// 1 / (B*H*W) = 2^-20, exact in fp32
#define INV_N 9.5367431640625e-7f

typedef float v2f __attribute__((ext_vector_type(2)));
typedef float v8f __attribute__((ext_vector_type(8)));

// ---------------------------------------------------------------------------
// Kernel 0: zero the scalar accumulator (required: d_out is poisoned, and we
// accumulate with atomics; runs first on `stream` each call -> deterministic).
// ---------------------------------------------------------------------------
__global__ void k_zero(float* __restrict__ out) {
    if (threadIdx.x == 0) out[0] = 0.0f;
}

// ---------------------------------------------------------------------------
// Kernel 1: vertical 1D EDT (fwd+bwd min scans) -> g^2 into workspace,
// plus per-sample bright-pixel count. One block per sample b; thread j owns
// column j, so global accesses are fully coalesced across the wave.
// ---------------------------------------------------------------------------
__global__ void k_vpass(const int* __restrict__ mask,
                        float* __restrict__ g2,
                        int* __restrict__ cnt) {
    const int b = blockIdx.x;
    const int j = threadIdx.x;
    const int*  mcol = mask + (size_t)b * H_ * W_ + j;
    float*      gcol = g2   + (size_t)b * H_ * W_ + j;

    int   nlocal = 0;
    float c = INF_;
    for (int i = 0; i < H_; ++i) {               // forward scan
        const int m = mcol[(size_t)i * W_];
        nlocal += m;
        const float v = m ? 0.0f : INF_;
        c = fminf(c + 1.0f, v);
        gcol[(size_t)i * W_] = c;
    }
    c = INF_;
    for (int i = H_ - 1; i >= 0; --i) {          // backward scan + square
        const int m = mcol[(size_t)i * W_];
        const float v = m ? 0.0f : INF_;
        c = fminf(c + 1.0f, v);
        const float g = fminf(c, gcol[(size_t)i * W_]);
        gcol[(size_t)i * W_] = g * g;
    }

    // block-reduce the bright-pixel count; one block per b -> plain store
    __shared__ int sc[W_];
    sc[j] = nlocal;
    __syncthreads();
    for (int s = W_ / 2; s > 0; s >>= 1) {
        if (j < s) sc[j] += sc[j + s];
        __syncthreads();
    }
    if (j == 0) cnt[b] = sc[0];
}

// ---------------------------------------------------------------------------
// Kernel 2: fused horizontal envelope + weight + cross-entropy + reduction.
// One block (256 threads) per (b, i) row. The k-scan reads the LDS-resident
// g^2 row with an all-lanes broadcast each step (bank-conflict free).
// Final reduction: per-wave row-sum via V_WMMA_F32_16X16X4_F32 with an
// all-ones B matrix (fp32 accumulate), shfl_xor(16) to fold the two M-halves,
// 8-entry LDS combine, one atomicAdd per block.
// ---------------------------------------------------------------------------
__global__ void k_loss(const float* __restrict__ pred,
                       const int*   __restrict__ mask,
                       const float* __restrict__ g2,
                       const int*   __restrict__ cnt,
                       float* __restrict__ out) {
    const int row = blockIdx.x;          // 0 .. B*H-1
    const int b   = row >> 8;            // row / H_
    const int i   = row & (H_ - 1);
    const int j   = threadIdx.x;

    // prefetch the prediction cachelines while g^2 row is staged into LDS
    const size_t p0 = (((size_t)b * 2 + 0) * H_ + i) * W_ + j;
    const size_t p1 = p0 + (size_t)H_ * W_;
    __builtin_prefetch(pred + p0, 0, 0);   // -> global_prefetch_b8
    __builtin_prefetch(pred + p1, 0, 0);

    __shared__ float sg[W_];
    sg[j] = g2[((size_t)b * H_ + i) * W_ + j];
    __syncthreads();

    // exact lower envelope: d2 = min_k g2[k] + (j-k)^2
    const float fj = (float)j;
    float best = 3.4e38f;
#pragma unroll 8
    for (int k = 0; k < W_; ++k) {
        const float dk = fj - (float)k;
        best = fminf(best, fmaf(dk, dk, sg[k]));
    }

    // distance weight (BETA = 2); zero if sample has no bright pixels
    const float w = (cnt[b] > 0) ? __expf(-0.5f * __fsqrt_rn(best)) : 0.0f;

    // 2-class CE from log-softmax
    const float x0 = pred[p0];
    const float x1 = pred[p1];
    const int   t  = mask[((size_t)b * H_ + i) * W_ + j];
    const float mx  = fmaxf(x0, x1);
    const float lse = mx + __logf(__expf(x0 - mx) + __expf(x1 - mx));
    const float ce  = lse - (t ? x1 : x0);

    const float val = ce * (1.0f + w) * INV_N;   // ALPHA = 1

    // ---- wave-level reduction via WMMA f32 16x16x4, B = ones ----
    // A: 2 fp32 per lane (64 slots = full 16x4 matrix); with B == 1 everywhere
    // D[m][n] = sum_k A[m][k] regardless of exact slot->element mapping.
    v2f a;    a.x = val;  a.y = 0.0f;
    v2f ones; ones.x = 1.0f; ones.y = 1.0f;
    v8f cacc = {};
    v8f d = __builtin_amdgcn_wmma_f32_16x16x4_f32(
        /*neg_a=*/false, a, /*neg_b=*/false, ones,
        /*c_mod=*/(short)0, cacc, /*reuse_a=*/false, /*reuse_b=*/false);

    float s = d[0] + d[1] + d[2] + d[3] + d[4] + d[5] + d[6] + d[7]; // M 0..7 half
    s += __shfl_xor(s, 16, 32);                                      // + other half

    __shared__ float sw[8];
    const int wid = j >> 5;
    if ((j & 31) == 0) sw[wid] = s;
    __syncthreads();
    if (j == 0) {
        const float tot = sw[0] + sw[1] + sw[2] + sw[3] +
                          sw[4] + sw[5] + sw[6] + sw[7];
        atomicAdd(out, tot);
    }
}

// ---------------------------------------------------------------------------
extern "C" void kernel_launch(void* const* d_in, const int* in_sizes, int n_in,
                              void* d_out, int out_size, void* d_ws, size_t ws_size,
                              hipStream_t stream) {
    const float* pred = (const float*)d_in[0];   // [16,2,256,256] f32
    const int*   mask = (const int*)d_in[1];     // [16,256,256] i32
    // d_in[2] (background_mask) is unused by the reference.
    float* out = (float*)d_out;                  // scalar f32

    float* g2  = (float*)d_ws;                                   // 4 MB scratch
    int*   cnt = (int*)((char*)d_ws + (size_t)B_ * H_ * W_ * sizeof(float));

    k_zero <<<1, 32, 0, stream>>>(out);
    k_vpass<<<B_, W_, 0, stream>>>(mask, g2, cnt);
    k_loss <<<B_ * H_, W_, 0, stream>>>(pred, mask, g2, cnt, out);
}